// GCNGMM_17772574671254
// MI455X (gfx1250) — compile-verified
//
#include <hip/hip_runtime.h>
#include <hip/hip_bf16.h>
#include <math.h>

#define N_NODES 100000
#define N_EDGES 1600000

typedef __attribute__((ext_vector_type(2))) float v2f;
typedef __attribute__((ext_vector_type(8))) float v8f;

// ---------------------------------------------------------------------------
// Degree / normalization
// ---------------------------------------------------------------------------
__global__ void k_fill_deg(float* __restrict__ deg, int n) {
    int i = blockIdx.x * blockDim.x + threadIdx.x;
    if (i < n) deg[i] = 1.0f;                       // self-loop
}

__global__ void k_accum_deg(const int* __restrict__ dst, float* __restrict__ deg, int e) {
    int i = blockIdx.x * blockDim.x + threadIdx.x;
    if (i < e) atomicAdd(&deg[dst[i]], 1.0f);
}

__global__ void k_to_dinv(float* __restrict__ deg, int n) {
    int i = blockIdx.x * blockDim.x + threadIdx.x;
    if (i < n) deg[i] = rsqrtf(deg[i]);             // in-place: deg -> dinv
}

// ---------------------------------------------------------------------------
// Layer 1 transform: (N,3) @ (3,16)  — too small for WMMA, plain FMA
// ---------------------------------------------------------------------------
__global__ void k_l1_transform(const float* __restrict__ x, const float* __restrict__ W1,
                               float* __restrict__ t, int n) {
    int idx = blockIdx.x * blockDim.x + threadIdx.x;
    if (idx >= n * 16) return;
    int node = idx >> 4, f = idx & 15;
    float x0 = x[node * 3 + 0], x1 = x[node * 3 + 1], x2 = x[node * 3 + 2];
    t[idx] = x0 * W1[f] + x1 * W1[16 + f] + x2 * W1[32 + f];
}

// ---------------------------------------------------------------------------
// WMMA GEMM: T[N,64] = H[N,K] @ W[K,64], fp32 via V_WMMA_F32_16X16X4_F32.
// One wave computes a 16x64 output tile; K stepped by 4.
// A layout (16x4 f32): lanes 0-15 -> {K=k,k+1}, lanes 16-31 -> {K=k+2,k+3}.
// B layout (4x16 f32): VGPR0 = rows k (lanes 0-15) / k+2 (lanes 16-31), VGPR1 = k+1 / k+3.
// D layout (16x16 f32): VGPR j -> row j (lanes 0-15) / row j+8 (lanes 16-31).
// ---------------------------------------------------------------------------
template <int K>
__global__ void k_gemm_wmma(const float* __restrict__ H, const float* __restrict__ W,
                            float* __restrict__ T, int nrows) {
    const int lane = threadIdx.x & 31;
    const int wave = blockIdx.x * (blockDim.x >> 5) + (threadIdx.x >> 5);
    const int row0 = wave * 16;
    if (row0 >= nrows) return;                      // wave-uniform: EXEC stays all-1s for WMMA
    const int half = lane >> 4;                     // 0: lanes 0-15, 1: lanes 16-31
    const int m    = lane & 15;
    const int r    = row0 + m;

    v8f acc0 = {}, acc1 = {}, acc2 = {}, acc3 = {};

    for (int k = 0; k < K; k += 4) {
        const int ka = k + 2 * half;
        v2f a;
        a.x = H[(size_t)r * K + ka];
        a.y = H[(size_t)r * K + ka + 1];
        const float* wr0 = W + (size_t)ka * 64;     // row ka
        const float* wr1 = wr0 + 64;                // row ka+1
        {
            v2f b; b.x = wr0[m];      b.y = wr1[m];
            acc0 = __builtin_amdgcn_wmma_f32_16x16x4_f32(false, a, false, b, (short)0, acc0, false, false);
        }
        {
            v2f b; b.x = wr0[16 + m]; b.y = wr1[16 + m];
            acc1 = __builtin_amdgcn_wmma_f32_16x16x4_f32(false, a, false, b, (short)0, acc1, false, false);
        }
        {
            v2f b; b.x = wr0[32 + m]; b.y = wr1[32 + m];
            acc2 = __builtin_amdgcn_wmma_f32_16x16x4_f32(false, a, false, b, (short)0, acc2, false, false);
        }
        {
            v2f b; b.x = wr0[48 + m]; b.y = wr1[48 + m];
            acc3 = __builtin_amdgcn_wmma_f32_16x16x4_f32(false, a, false, b, (short)0, acc3, false, false);
        }
    }

    float* out = T + (size_t)row0 * 64;
    #pragma unroll
    for (int j = 0; j < 8; ++j) {
        const int rr = j + 8 * half;
        out[(size_t)rr * 64 +  0 + m] = acc0[j];
        out[(size_t)rr * 64 + 16 + m] = acc1[j];
        out[(size_t)rr * 64 + 32 + m] = acc2[j];
        out[(size_t)rr * 64 + 48 + m] = acc3[j];
    }
}

// ---------------------------------------------------------------------------
// agg = snorm * t + b  (self-loop term + bias; message atomics add on top)
// ---------------------------------------------------------------------------
__global__ void k_agg_init(const float* __restrict__ t, const float* __restrict__ dinv,
                           const float* __restrict__ b, float* __restrict__ agg,
                           int n, int F) {
    int idx = blockIdx.x * blockDim.x + threadIdx.x;
    if (idx >= n * F) return;
    int node = idx / F, f = idx - node * F;
    float di = dinv[node];
    agg[idx] = di * di * t[idx] + b[f];
}

// ---------------------------------------------------------------------------
// Edge scatter, F=16: thread per (edge, feature)
// ---------------------------------------------------------------------------
__global__ void k_edge16(const int* __restrict__ src, const int* __restrict__ dst,
                         const float* __restrict__ dinv, const float* __restrict__ t,
                         float* __restrict__ agg, int e) {
    int idx = blockIdx.x * blockDim.x + threadIdx.x;
    if (idx >= e * 16) return;
    int ed = idx >> 4, f = idx & 15;
    int s = src[ed], d = dst[ed];
    float en = dinv[s] * dinv[d];
    atomicAdd(&agg[(size_t)d * 16 + f], en * t[(size_t)s * 16 + f]);
}

// ---------------------------------------------------------------------------
// Edge scatter, F=64: wave per edge-group, lane = feature pair (float2, L2-coalesced)
// ---------------------------------------------------------------------------
#define EPW 4
__global__ void k_edge64(const int* __restrict__ src, const int* __restrict__ dst,
                         const float* __restrict__ dinv, const float* __restrict__ t,
                         float* __restrict__ agg, int e) {
    const int lane = threadIdx.x & 31;
    const int wave = blockIdx.x * (blockDim.x >> 5) + (threadIdx.x >> 5);
    const int e0 = wave * EPW;
    for (int i = 0; i < EPW; ++i) {
        int ed = e0 + i;
        if (ed >= e) return;
        int s = src[ed], d = dst[ed];
        float en = dinv[s] * dinv[d];
        const float2 v = *(const float2*)(t + (size_t)s * 64 + lane * 2);
        float* ap = agg + (size_t)d * 64 + lane * 2;
        atomicAdd(ap,     en * v.x);
        atomicAdd(ap + 1, en * v.y);
    }
}

__global__ void k_relu(float* __restrict__ h, int total) {
    int i = blockIdx.x * blockDim.x + threadIdx.x;
    if (i < total) h[i] = fmaxf(h[i], 0.0f);
}

// ---------------------------------------------------------------------------
// 1/||p||  (p has 64 elements; single wave32)
// ---------------------------------------------------------------------------
__global__ void k_pnorm(const float* __restrict__ p, float* __restrict__ pinv) {
    int lane = threadIdx.x;
    float a = p[lane], b = p[lane + 32];
    float s = a * a + b * b;
    for (int off = 16; off > 0; off >>= 1) s += __shfl_down(s, off, 32);
    if (lane == 0) pinv[0] = 1.0f / sqrtf(s);
}

__global__ void k_score(const float* __restrict__ h, const float* __restrict__ p,
                        const float* __restrict__ pinv, float* __restrict__ score, int n) {
    int i = blockIdx.x * blockDim.x + threadIdx.x;
    if (i >= n) return;
    const float4* hr = (const float4*)(h + (size_t)i * 64);
    const float4* pv = (const float4*)p;
    float s = 0.0f;
    #pragma unroll
    for (int k = 0; k < 16; ++k) {
        float4 av = hr[k], bv = pv[k];
        s += av.x * bv.x + av.y * bv.y + av.z * bv.z + av.w * bv.w;
    }
    score[i] = s * pinv[0];
}

// ---------------------------------------------------------------------------
// Top-10 (jax.lax.top_k semantics: descending, ties -> lower index) + heads.
// Single 1024-thread block: 10 masked-argmax rounds, then tiny GEMV heads.
// Output: mean[10,2] | log_std[10,2] | w[10,1]==1  -> 50 floats flat.
// ---------------------------------------------------------------------------
__global__ void k_topk_heads(const float* __restrict__ score, const float* __restrict__ h,
                             const float* __restrict__ Wm, const float* __restrict__ bm,
                             const float* __restrict__ Ws, const float* __restrict__ bs,
                             float* __restrict__ out, int n) {
    __shared__ float sval[1024];
    __shared__ int   sidx[1024];
    __shared__ int   chosen[10];
    __shared__ float chval[10];
    __shared__ float hk[10][64];
    const int tid = threadIdx.x;

    for (int r = 0; r < 10; ++r) {
        int lc[10];
        for (int c = 0; c < r; ++c) lc[c] = chosen[c];
        float bv = -INFINITY;
        int   bi = 0x7fffffff;
        for (int i = tid; i < n; i += 1024) {
            bool skip = false;
            for (int c = 0; c < r; ++c) skip |= (lc[c] == i);
            if (skip) continue;
            float v = score[i];
            if (v > bv || (v == bv && i < bi)) { bv = v; bi = i; }
        }
        sval[tid] = bv; sidx[tid] = bi;
        __syncthreads();
        for (int s = 512; s > 0; s >>= 1) {
            if (tid < s) {
                float v2 = sval[tid + s]; int i2 = sidx[tid + s];
                if (v2 > sval[tid] || (v2 == sval[tid] && i2 < sidx[tid])) {
                    sval[tid] = v2; sidx[tid] = i2;
                }
            }
            __syncthreads();
        }
        if (tid == 0) { chosen[r] = sidx[0]; chval[r] = sval[0]; }
        __syncthreads();
    }

    if (tid < 640) {
        int i = tid >> 6, k = tid & 63;
        hk[i][k] = h[(size_t)chosen[i] * 64 + k] * tanhf(chval[i]);
    }
    __syncthreads();

    if (tid < 20) {                                  // mean = hk @ Wm + bm
        int i = tid >> 1, j = tid & 1;
        float s = bm[j];
        for (int k = 0; k < 64; ++k) s += hk[i][k] * Wm[k * 2 + j];
        out[tid] = s;
    } else if (tid < 40) {                           // log_std = hk @ Ws + bs
        int t2 = tid - 20; int i = t2 >> 1, j = t2 & 1;
        float s = bs[j];
        for (int k = 0; k < 64; ++k) s += hk[i][k] * Ws[k * 2 + j];
        out[20 + t2] = s;
    } else if (tid < 50) {                           // softmax over singleton dim -> 1
        out[tid] = 1.0f;
    }
}

// ---------------------------------------------------------------------------
// Launch
// ---------------------------------------------------------------------------
extern "C" void kernel_launch(void* const* d_in, const int* in_sizes, int n_in,
                              void* d_out, int out_size, void* d_ws, size_t ws_size,
                              hipStream_t stream) {
    const float* x   = (const float*)d_in[0];
    const int*   ei  = (const int*)d_in[1];
    const float* W1  = (const float*)d_in[2];
    const float* b1  = (const float*)d_in[3];
    const float* W2  = (const float*)d_in[4];
    const float* b2  = (const float*)d_in[5];
    const float* W3  = (const float*)d_in[6];
    const float* b3  = (const float*)d_in[7];
    const float* p   = (const float*)d_in[8];
    const float* Wm  = (const float*)d_in[9];
    const float* bm  = (const float*)d_in[10];
    const float* Ws  = (const float*)d_in[11];
    const float* bs  = (const float*)d_in[12];
    float* out = (float*)d_out;

    const int N = N_NODES, E = N_EDGES;
    const int* src = ei;
    const int* dst = ei + E;

    float* wsf    = (float*)d_ws;
    float* dinv   = wsf;                              // N
    float* bufT   = wsf + N;                          // N*64 (transform output)
    float* bufH   = bufT + (size_t)N * 64;            // N*64 (agg -> h, in place)
    float* scoreb = bufH + (size_t)N * 64;            // N
    float* pinv   = scoreb + N;                       // 1

    const int B = 256;
    const int gN    = (N + B - 1) / B;
    const int gE    = (E + B - 1) / B;
    const int gN16  = (N * 16 + B - 1) / B;
    const int gN64  = (N * 64 + B - 1) / B;
    const int gE16  = (E * 16 + B - 1) / B;
    const int gE64  = (E + 8 * EPW - 1) / (8 * EPW);  // 8 waves/block * EPW edges/wave
    const int tiles = N / 16;                         // 100000 / 16 = 6250 exact
    const int gGemm = (tiles + 7) / 8;                // 8 waves per 256-thread block

    // Normalization
    k_fill_deg<<<gN, B, 0, stream>>>(dinv, N);
    k_accum_deg<<<gE, B, 0, stream>>>(dst, dinv, E);
    k_to_dinv<<<gN, B, 0, stream>>>(dinv, N);

    // Layer 1: 3 -> 16
    k_l1_transform<<<gN16, B, 0, stream>>>(x, W1, bufT, N);
    k_agg_init<<<gN16, B, 0, stream>>>(bufT, dinv, b1, bufH, N, 16);
    k_edge16<<<gE16, B, 0, stream>>>(src, dst, dinv, bufT, bufH, E);
    k_relu<<<gN16, B, 0, stream>>>(bufH, N * 16);

    // Layer 2: 16 -> 64 (WMMA)
    k_gemm_wmma<16><<<gGemm, B, 0, stream>>>(bufH, W2, bufT, N);
    k_agg_init<<<gN64, B, 0, stream>>>(bufT, dinv, b2, bufH, N, 64);
    k_edge64<<<gE64, B, 0, stream>>>(src, dst, dinv, bufT, bufH, E);
    k_relu<<<gN64, B, 0, stream>>>(bufH, N * 64);

    // Layer 3: 64 -> 64 (WMMA)
    k_gemm_wmma<64><<<gGemm, B, 0, stream>>>(bufH, W3, bufT, N);
    k_agg_init<<<gN64, B, 0, stream>>>(bufT, dinv, b3, bufH, N, 64);
    k_edge64<<<gE64, B, 0, stream>>>(src, dst, dinv, bufT, bufH, E);
    k_relu<<<gN64, B, 0, stream>>>(bufH, N * 64);

    // TopK pooling + GMM heads
    k_pnorm<<<1, 32, 0, stream>>>(p, pinv);
    k_score<<<gN, B, 0, stream>>>(bufH, p, pinv, scoreb, N);
    k_topk_heads<<<1, 1024, 0, stream>>>(scoreb, bufH, Wm, bm, Ws, bs, out, N);
}